// GCMC_4269197492538
// MI455X (gfx1250) — compile-verified
//
#include <hip/hip_runtime.h>

#define N_USERS 60000
#define N_ITEMS 40000
#define NNODES  (N_USERS + N_ITEMS)   // 100000
#define EMB     128
#define NEDGES  1600000
#define NTILES_M ((NNODES + 15) / 16) // 6250

typedef __attribute__((ext_vector_type(2))) float v2f;
typedef __attribute__((ext_vector_type(8))) float v8f;

// ---------------------------------------------------------------------------
// emb = concat(user_emb, item_emb); total(d_out) = emb; x = emb
// ---------------------------------------------------------------------------
__global__ void init_concat(const float* __restrict__ ue,
                            const float* __restrict__ ie,
                            float* __restrict__ x,
                            float* __restrict__ out) {
  long long i = (long long)blockIdx.x * blockDim.x + threadIdx.x;
  const long long nu = (long long)N_USERS * EMB;
  float v = (i < nu) ? ue[i] : ie[i - nu];
  x[i] = v;
  out[i] = v;
}

__global__ void zero_buf(float* __restrict__ y) {
  long long i = (long long)blockIdx.x * blockDim.x + threadIdx.x;
  y[i] = 0.0f;
}

// ---------------------------------------------------------------------------
// SpMM scatter: y[row[e]] += val[e] * x[col[e]]
// One wave32 per edge; each lane handles a float4 chunk (128 = 32 lanes * 4).
// Row reads are 512B contiguous per edge -> fully coalesced; scatter uses
// native non-returning global_atomic_add_f32 (relaxed, agent scope).
// ---------------------------------------------------------------------------
__global__ void spmm_atomic(const int*   __restrict__ rowi,
                            const int*   __restrict__ coli,
                            const float* __restrict__ val,
                            const float* __restrict__ x,
                            float*       __restrict__ y) {
  long long t = (long long)blockIdx.x * blockDim.x + threadIdx.x;
  int e    = (int)(t >> 5);
  int lane = (int)(t & 31);
  if (e >= NEDGES) return;
  int   r = rowi[e];
  int   c = coli[e];
  float w = val[e];
  const float4 xv = *(const float4*)(x + (long long)c * EMB + lane * 4);
  float* yp = y + (long long)r * EMB + lane * 4;
  __hip_atomic_fetch_add(yp + 0, w * xv.x, __ATOMIC_RELAXED, __HIP_MEMORY_SCOPE_AGENT);
  __hip_atomic_fetch_add(yp + 1, w * xv.y, __ATOMIC_RELAXED, __HIP_MEMORY_SCOPE_AGENT);
  __hip_atomic_fetch_add(yp + 2, w * xv.z, __ATOMIC_RELAXED, __HIP_MEMORY_SCOPE_AGENT);
  __hip_atomic_fetch_add(yp + 3, w * xv.w, __ATOMIC_RELAXED, __HIP_MEMORY_SCOPE_AGENT);
}

// ---------------------------------------------------------------------------
// Z = relu(A @ W), A: NNODES x 128, W: 128 x 128 (f32).
// V_WMMA_F32_16X16X4_F32: one wave computes a 16x16 tile, K-loop of 32 WMMAs.
// Block = 8 waves = 8 row tiles sharing one 128x16 column panel of W in LDS.
// A-fragment (16x4 f32, ISA 7.12.2): lanes 0-15 -> K={k,k+1}, lanes 16-31 ->
// K={k+2,k+3}, M = lane&15.  B (4x16): rows striped across lanes per VGPR.
// C/D (16x16 f32, 8 VGPRs): VGPR r -> M = r + (lane>=16 ? 8 : 0), N = lane&15.
// ---------------------------------------------------------------------------
__global__ __launch_bounds__(256) void gemm_relu_wmma(
    const float* __restrict__ A,
    const float* __restrict__ W,
    float*       __restrict__ Z) {
  __shared__ float Wl[EMB * 16];          // 8 KB column panel of W
  const int colBase = blockIdx.x * 16;

  for (int i = threadIdx.x; i < EMB * 16; i += blockDim.x) {
    int k = i >> 4, n = i & 15;
    Wl[i] = W[k * EMB + colBase + n];
  }
  __syncthreads();

  const int wave    = threadIdx.x >> 5;
  const int lane    = threadIdx.x & 31;
  const int rowTile = blockIdx.y * 8 + wave;
  if (rowTile >= NTILES_M) return;

  const int m      = lane & 15;           // M (for A) / N (for B,D)
  const int kOff   = (lane >> 4) * 2;     // 0 or 2
  const float* arow = A + (long long)(rowTile * 16 + m) * EMB;

  v8f acc = {};
#pragma unroll
  for (int k = 0; k < EMB; k += 4) {
    v2f a = *(const v2f*)(arow + k + kOff);              // K = k+kOff, k+kOff+1
    v2f b;
    b.x = Wl[(k + kOff    ) * 16 + m];
    b.y = Wl[(k + kOff + 1) * 16 + m];
    acc = __builtin_amdgcn_wmma_f32_16x16x4_f32(
        /*neg_a=*/false, a, /*neg_b=*/false, b,
        /*c_mod=*/(short)0, acc, /*reuse_a=*/false, /*reuse_b=*/false);
  }

  const int col   = colBase + m;
  const int rbase = rowTile * 16 + (lane >> 4) * 8;
#pragma unroll
  for (int r = 0; r < 8; ++r) {
    float v = fmaxf(acc[r], 0.0f);                       // fused ReLU
    Z[(long long)(rbase + r) * EMB + col] = v;
  }
}

// ---------------------------------------------------------------------------
// Row L2-normalize x in place and accumulate into total (d_out).
// One wave32 per row; __shfl_xor tree reduce (wave32-native).
// ---------------------------------------------------------------------------
__global__ void norm_accum(float* __restrict__ x, float* __restrict__ out) {
  long long t = (long long)blockIdx.x * blockDim.x + threadIdx.x;
  int rowIdx = (int)(t >> 5);
  int lane   = (int)(t & 31);
  if (rowIdx >= NNODES) return;
  float4* xp = (float4*)(x + (long long)rowIdx * EMB + lane * 4);
  float4 v = *xp;
  float ss = v.x * v.x + v.y * v.y + v.z * v.z + v.w * v.w;
#pragma unroll
  for (int off = 16; off > 0; off >>= 1) ss += __shfl_xor(ss, off, 32);
  float scale = 1.0f / fmaxf(sqrtf(ss), 1e-12f);
  v.x *= scale; v.y *= scale; v.z *= scale; v.w *= scale;
  *xp = v;
  float4* op = (float4*)(out + (long long)rowIdx * EMB + lane * 4);
  float4 o = *op;
  o.x += v.x; o.y += v.y; o.z += v.z; o.w += v.w;
  *op = o;
}

// ---------------------------------------------------------------------------
extern "C" void kernel_launch(void* const* d_in, const int* in_sizes, int n_in,
                              void* d_out, int out_size, void* d_ws, size_t ws_size,
                              hipStream_t stream) {
  const int*   adj_row = (const int*)d_in[0];
  const int*   adj_col = (const int*)d_in[1];
  const float* adj_val = (const float*)d_in[2];
  const float* ue      = (const float*)d_in[3];
  const float* ie      = (const float*)d_in[4];
  const float* Ws[3]   = {(const float*)d_in[5], (const float*)d_in[6],
                          (const float*)d_in[7]};
  float* out = (float*)d_out;

  float* x = (float*)d_ws;                        // 100000 x 128 ping buffer
  float* y = x + (long long)NNODES * EMB;         // 100000 x 128 SpMM target

  const int ELEMS = NNODES * EMB;                 // 12,800,000 (exact /256)
  init_concat<<<ELEMS / 256, 256, 0, stream>>>(ue, ie, x, out);

  for (int l = 0; l < 3; ++l) {
    zero_buf<<<ELEMS / 256, 256, 0, stream>>>(y);
    spmm_atomic<<<(NEDGES * 32) / 256, 256, 0, stream>>>(adj_row, adj_col,
                                                         adj_val, x, y);
    dim3 grid(EMB / 16, (NTILES_M + 7) / 8);      // 8 x 782
    gemm_relu_wmma<<<grid, 256, 0, stream>>>(y, Ws[l], x);
    norm_accum<<<(NNODES * 32) / 256, 256, 0, stream>>>(x, out);
  }
}